// SynAlign_29867202576855
// MI455X (gfx1250) — compile-verified
//
#include <hip/hip_runtime.h>

typedef __attribute__((ext_vector_type(2))) float v2f;
typedef __attribute__((ext_vector_type(8))) float v8f;

#define B_ 256
#define L_ 128
#define D_ 1024
static constexpr float NEG_INF = -999.0f;

// d_out flat offsets (floats), in reference return order:
// (s, source_att, at_soft, t, target_att, ta_soft)
static constexpr size_t OFF_S    = 0;
static constexpr size_t OFF_SATT = 33554432;   // B*L*D
static constexpr size_t OFF_ATS  = 67108864;   // + B*L*D
static constexpr size_t OFF_T    = 71303168;   // + B*L*L
static constexpr size_t OFF_TATT = 104857600;  // + B*L*D
static constexpr size_t OFF_TAS  = 138412032;  // + B*L*D

// ---------------------------------------------------------------------------
// Kernel 1: embedding gather + 3-tap average pool (TF SAME, edge divisor 2)
// grid: (B*L, 2)  block: 256 threads, each thread handles one float4 of D
// ---------------------------------------------------------------------------
__global__ __launch_bounds__(256) void gather_pool_kernel(
    const int* __restrict__ ssent, const int* __restrict__ tsent,
    const float* __restrict__ semb, const float* __restrict__ temb,
    float* __restrict__ out_s, float* __restrict__ out_t) {
  const int row = blockIdx.x;            // b*L + l
  const int which = blockIdx.y;          // 0 = source, 1 = target
  const int b = row >> 7;
  const int l = row & (L_ - 1);
  const int* __restrict__ sent = which ? tsent : ssent;
  const float* __restrict__ emb = which ? temb : semb;
  float* __restrict__ outp = which ? out_t : out_s;

  const int d = threadIdx.x << 2;        // 0..1020 step 4

  const int idc = sent[b * L_ + l];
  float4 acc = *(const float4*)(emb + (size_t)idc * D_ + d);
  float cnt = 1.0f;
  if (l > 0) {
    const int idl = sent[b * L_ + l - 1];
    const float4 v = *(const float4*)(emb + (size_t)idl * D_ + d);
    acc.x += v.x; acc.y += v.y; acc.z += v.z; acc.w += v.w;
    cnt += 1.0f;
  }
  if (l < L_ - 1) {
    const int idr = sent[b * L_ + l + 1];
    const float4 v = *(const float4*)(emb + (size_t)idr * D_ + d);
    acc.x += v.x; acc.y += v.y; acc.z += v.z; acc.w += v.w;
    cnt += 1.0f;
  }
  const float inv = 1.0f / cnt;
  acc.x *= inv; acc.y *= inv; acc.z *= inv; acc.w *= inv;
  *(float4*)(outp + (size_t)row * D_ + d) = acc;
}

// ---------------------------------------------------------------------------
// Kernel 2: masked score GEMM  score[b,t,s] = sum_d t[b,t,d]*s[b,s,d]
// One wave per 16x16 tile; f32 WMMA 16x16x4 over D=1024.
// Uniform scalar K-loop (lane offset folded into pointers) so no EXEC games;
// two interleaved accumulator chains for XDL ILP; branchless mask epilogue
// (8 contiguous tmask bytes fetched as one u64 before the loop).
// grid: (B, 8)  block: 256 (8 waves -> 64 tiles per batch)
// ---------------------------------------------------------------------------
__global__ __launch_bounds__(256) void score_kernel(
    const float* __restrict__ smat, const float* __restrict__ tmat,
    const unsigned char* __restrict__ smask, const unsigned char* __restrict__ tmask,
    float* __restrict__ score) {
  const int b = blockIdx.x;
  const int wave = threadIdx.x >> 5;
  const int lane = threadIdx.x & 31;
  const int task = blockIdx.y * 8 + wave;          // 0..63
  const int m_base = (task >> 3) << 4;             // t-tile row base (M)
  const int n_base = (task & 7) << 4;              // s-tile row base (N)
  const int half = lane >> 4;
  const int lm = lane & 15;

  // Per-lane mask bits, fetched up front (overlap with the K-loop).
  const int sg = n_base + lm;                      // global s index (N)
  const unsigned int sv = smask[b * L_ + sg];      // 0 or 1
  const unsigned long long t8 =                    // tmask[m_base+8*half .. +7]
      *(const unsigned long long*)(tmask + b * L_ + m_base + 8 * half);

  const float* __restrict__ tp = tmat + ((size_t)b * L_ + m_base + lm) * D_ + 2 * half;
  const float* __restrict__ sp = smat + ((size_t)b * L_ + n_base + lm) * D_ + 2 * half;

  const v8f vzero = {0.f, 0.f, 0.f, 0.f, 0.f, 0.f, 0.f, 0.f};
  v8f acc0 = vzero, acc1 = vzero;
#pragma unroll 4
  for (int i = 0; i < D_ / 8; ++i) {               // uniform trip count: 128
    const v2f a0 = *(const v2f*)(tp);
    const v2f b0 = *(const v2f*)(sp);
    const v2f a1 = *(const v2f*)(tp + 4);
    const v2f b1 = *(const v2f*)(sp + 4);
    acc0 = __builtin_amdgcn_wmma_f32_16x16x4_f32(false, a0, false, b0,
                                                 (short)0, acc0, false, false);
    acc1 = __builtin_amdgcn_wmma_f32_16x16x4_f32(false, a1, false, b1,
                                                 (short)0, acc1, false, false);
    tp += 8; sp += 8;
  }

  // C/D layout: N = lane%16, M = r + 8*(lane/16) for VGPR r.
  float* __restrict__ orow = score + ((size_t)b * L_ + m_base + 8 * half) * L_ + sg;
#pragma unroll
  for (int r = 0; r < 8; ++r) {
    const unsigned int tv = (unsigned int)((t8 >> (8 * r)) & 0xffull);
    const float val = acc0[r] + acc1[r];
    orow[(size_t)r * L_] = (sv & tv) ? val : NEG_INF;   // v_cndmask, no branch
  }
}

// ---------------------------------------------------------------------------
// Kernel 3: dual softmax over the masked score matrix.
//  grid.y==0: ta_soft[b,t,:] = softmax_s score[b,t,s]   (row-contig)
//  grid.y==1: at_soft[b,s,:] = softmax_t score[b,t,s]   (column reads)
// One wave per output row; lane holds 4 elements; __shfl_xor reductions.
// grid: (B*L/8, 2)  block: 256
// ---------------------------------------------------------------------------
__global__ __launch_bounds__(256) void softmax_kernel(
    const float* __restrict__ score,
    float* __restrict__ ta_soft, float* __restrict__ at_soft) {
  const int wave = threadIdx.x >> 5;
  const int lane = threadIdx.x & 31;
  const int row = blockIdx.x * 8 + wave;           // 0..B*L-1
  const int b = row >> 7;
  const int r = row & (L_ - 1);

  float v[4];
  if (blockIdx.y == 0) {
    const float4 x = *(const float4*)(score + ((size_t)b * L_ + r) * L_ + 4 * lane);
    v[0] = x.x; v[1] = x.y; v[2] = x.z; v[3] = x.w;
  } else {
#pragma unroll
    for (int j = 0; j < 4; ++j)
      v[j] = score[((size_t)b * L_ + 4 * lane + j) * L_ + r];
  }

  float m = fmaxf(fmaxf(v[0], v[1]), fmaxf(v[2], v[3]));
#pragma unroll
  for (int i = 16; i >= 1; i >>= 1) m = fmaxf(m, __shfl_xor(m, i, 32));

  float sum = 0.f;
#pragma unroll
  for (int j = 0; j < 4; ++j) { v[j] = __expf(v[j] - m); sum += v[j]; }
#pragma unroll
  for (int i = 16; i >= 1; i >>= 1) sum += __shfl_xor(sum, i, 32);

  const float inv = 1.0f / sum;
  float4 o;
  o.x = v[0] * inv; o.y = v[1] * inv; o.z = v[2] * inv; o.w = v[3] * inv;
  float* __restrict__ outp = (blockIdx.y == 0) ? ta_soft : at_soft;
  *(float4*)(outp + (size_t)row * L_ + 4 * lane) = o;
}

// ---------------------------------------------------------------------------
// Kernel 4: attention GEMMs
//   source_att = ta_soft[b,T,S] @ s[b,S,D];  target_att = at_soft @ t
// One wave computes a 16x64 strip (1 M-tile x 4 N-tiles); A-frag reused
// across the 4 N-tiles; 4 independent accumulator chains give XDL ILP;
// B loads are 16-lane-coalesced (64B per half-wave) along D.
// grid: (B, 32)  block: 256  (tasks 0..127 -> source_att, 128..255 -> target)
// ---------------------------------------------------------------------------
__global__ __launch_bounds__(256) void attn_kernel(
    const float* __restrict__ smat, const float* __restrict__ tmat,
    const float* __restrict__ ta_soft, const float* __restrict__ at_soft,
    float* __restrict__ satt, float* __restrict__ tatt) {
  const int b = blockIdx.x;
  const int wave = threadIdx.x >> 5;
  const int lane = threadIdx.x & 31;
  const int task = blockIdx.y * 8 + wave;          // 0..255
  const int which = task >> 7;                     // 0: source_att, 1: target_att
  const int sub = task & 127;
  const int m_base = (sub >> 4) << 4;              // 8 M tiles
  const int n0 = (sub & 15) * 64;                  // 16 N-groups of 64 cols
  const int half = lane >> 4;
  const int lm = lane & 15;

  const float* __restrict__ P = which ? at_soft : ta_soft;   // [128 x 128] probs
  const float* __restrict__ Vm = which ? tmat : smat;        // [128 x 1024] values
  float* __restrict__ O = which ? tatt : satt;

  const float* __restrict__ prow =
      P + ((size_t)b * L_ + m_base + lm) * L_ + 2 * half;
  const float* __restrict__ vcol =
      Vm + (size_t)b * L_ * D_ + (size_t)(2 * half) * D_ + n0 + lm;

  const v8f vzero = {0.f, 0.f, 0.f, 0.f, 0.f, 0.f, 0.f, 0.f};
  v8f acc[4];
#pragma unroll
  for (int j = 0; j < 4; ++j) acc[j] = vzero;

#pragma unroll 2
  for (int i = 0; i < L_ / 4; ++i) {               // uniform trip count: 32
    const v2f a = *(const v2f*)(prow);             // A-frag: P[m, ka..ka+1]
#pragma unroll
    for (int j = 0; j < 4; ++j) {
      v2f bb;                                      // B-frag: Vm[ka+v, n]
      bb.x = vcol[j * 16];
      bb.y = vcol[j * 16 + D_];
      acc[j] = __builtin_amdgcn_wmma_f32_16x16x4_f32(
          false, a, false, bb, (short)0, acc[j], false, false);
    }
    prow += 4;
    vcol += (size_t)4 * D_;
  }

  float* __restrict__ obase =
      O + ((size_t)b * L_ + m_base + 8 * half) * D_ + n0 + lm;
#pragma unroll
  for (int j = 0; j < 4; ++j) {
#pragma unroll
    for (int r = 0; r < 8; ++r) {
      obase[(size_t)r * D_ + j * 16] = acc[j][r];
    }
  }
}

// ---------------------------------------------------------------------------
extern "C" void kernel_launch(void* const* d_in, const int* in_sizes, int n_in,
                              void* d_out, int out_size, void* d_ws, size_t ws_size,
                              hipStream_t stream) {
  (void)in_sizes; (void)n_in; (void)out_size; (void)ws_size;
  const int* ssent = (const int*)d_in[0];
  const int* tsent = (const int*)d_in[1];
  const unsigned char* smask = (const unsigned char*)d_in[2];
  const unsigned char* tmask = (const unsigned char*)d_in[3];
  const float* semb = (const float*)d_in[4];
  const float* temb = (const float*)d_in[5];

  float* out = (float*)d_out;
  float* s_out  = out + OFF_S;
  float* satt   = out + OFF_SATT;
  float* at     = out + OFF_ATS;
  float* t_out  = out + OFF_T;
  float* tatt   = out + OFF_TATT;
  float* ta     = out + OFF_TAS;

  float* score = (float*)d_ws;   // B*L*L f32 = 16 MB masked score matrix

  gather_pool_kernel<<<dim3(B_ * L_, 2), 256, 0, stream>>>(
      ssent, tsent, semb, temb, s_out, t_out);

  score_kernel<<<dim3(B_, 8), 256, 0, stream>>>(
      s_out, t_out, smask, tmask, score);

  softmax_kernel<<<dim3(B_ * L_ / 8, 2), 256, 0, stream>>>(score, ta, at);

  attn_kernel<<<dim3(B_, 32), 256, 0, stream>>>(
      s_out, t_out, ta, at, satt, tatt);
}